// GraphConvolution_79388175499516
// MI455X (gfx1250) — compile-verified
//
#include <hip/hip_runtime.h>

// ---------------------------------------------------------------------------
// GCN layer for MI455X (gfx1250, wave32).
//   out = relu( D~^-1/2 (A+I) D~^-1/2 @ feat @ W + b )
// Pipeline:
//   k1: deg = 0
//   k2: deg[dst[e]] += 1                (fp32 atomics, exact for counts)
//   k3: dinv = rsqrt(deg+1); agg = feat * dinv^2   (self-loop fused into init)
//   k4: agg[dst] += feat[src] * dinv[src]*dinv[dst]  (wave per edge, float2/lane)
//   k5: out = relu(agg @ W + b) via V_WMMA_F32_16X16X4_F32 (f32 matrix core)
//       with W staged in LDS in B-fragment (K-pair interleaved) layout so each
//       B fragment is a single ds_load_b64 into an aligned VGPR pair.
// ---------------------------------------------------------------------------

typedef float v2f __attribute__((ext_vector_type(2)));
typedef float v8f __attribute__((ext_vector_type(8)));

#define F 64  // F_IN == F_OUT == 64

__global__ __launch_bounds__(256) void gcn_zero_deg(float* __restrict__ deg, int N) {
    int i = blockIdx.x * 256 + threadIdx.x;
    if (i < N) deg[i] = 0.0f;
}

__global__ __launch_bounds__(256) void gcn_count_deg(const int* __restrict__ dst,
                                                     float* __restrict__ deg, int E) {
    int e = blockIdx.x * 256 + threadIdx.x;
    if (e < E) atomicAdd(&deg[dst[e]], 1.0f);
}

// wave per node: dinv[i] = rsqrt(deg[i]+1); agg[i][:] = feat[i][:] * dinv[i]^2
__global__ __launch_bounds__(256) void gcn_selfloop(const float* __restrict__ feat,
                                                    const float* __restrict__ deg,
                                                    float* __restrict__ dinv,
                                                    float* __restrict__ agg, int N) {
    int node = blockIdx.x * 8 + (threadIdx.x >> 5);
    if (node >= N) return;
    int lane = threadIdx.x & 31;
    float di = __frsqrt_rn(deg[node] + 1.0f);
    if (lane == 0) dinv[node] = di;
    float s = di * di;
    v2f f = *(const v2f*)(feat + (size_t)node * F + lane * 2);
    v2f r; r.x = f.x * s; r.y = f.y * s;
    *(v2f*)(agg + (size_t)node * F + lane * 2) = r;
}

// wave per edge: agg[dst] += feat[src] * dinv[src]*dinv[dst]
__global__ __launch_bounds__(256) void gcn_scatter(const float* __restrict__ feat,
                                                   const int* __restrict__ src,
                                                   const int* __restrict__ dst,
                                                   const float* __restrict__ dinv,
                                                   float* __restrict__ agg, int E) {
    int e = blockIdx.x * 8 + (threadIdx.x >> 5);
    if (e >= E) return;                 // uniform within wave
    int lane = threadIdx.x & 31;
    int s = src[e];
    int d = dst[e];
    float w = dinv[s] * dinv[d];
    v2f f = *(const v2f*)(feat + (size_t)s * F + lane * 2);
    float* p = agg + (size_t)d * F + lane * 2;
    atomicAdd(p + 0, f.x * w);          // global_atomic_add_f32
    atomicAdd(p + 1, f.y * w);
}

// 8 waves / block; each wave computes a 16-row x 64-col output tile with
// V_WMMA_F32_16X16X4_F32 (A 16x4, B 4x16, C 16x16 per ISA lane layouts).
__global__ __launch_bounds__(256) void gcn_gemm_relu(const float* __restrict__ agg,
                                                     const float* __restrict__ W,
                                                     const float* __restrict__ bias,
                                                     float* __restrict__ out, int N) {
    // B-fragment layout: sW2[p*F + n] = {W[2p][n], W[2p+1][n]}, p = 0..31
    __shared__ v2f   sW2[(F / 2) * F];  // 16 KB
    __shared__ float sB[F];

    for (int i = threadIdx.x; i < (F / 2) * F; i += 256) {
        int p = i >> 6;                 // K-pair index
        int n = i & 63;                 // column
        v2f t;
        t.x = W[(2 * p + 0) * F + n];
        t.y = W[(2 * p + 1) * F + n];
        sW2[i] = t;
    }
    if (threadIdx.x < F) sB[threadIdx.x] = bias[threadIdx.x];
    __syncthreads();

    int wave = threadIdx.x >> 5;
    int lane = threadIdx.x & 31;
    int rowBase = (blockIdx.x * 8 + wave) * 16;
    if (rowBase >= N) return;           // uniform per wave

    int half = lane >> 4;               // 0: K=k,k+1 / rows M0-7 ; 1: K=k+2,k+3 / rows M8-15
    int l16  = lane & 15;
    int koff = half * 2;

    // A-row for this lane (clamped so EXEC stays all-ones for WMMA)
    int arow = rowBase + l16;
    if (arow >= N) arow = N - 1;
    const float* aptr = agg + (size_t)arow * F;

    v8f acc0 = {}, acc1 = {}, acc2 = {}, acc3 = {};

#pragma unroll
    for (int k = 0; k < F; k += 4) {
        // A 16x4 fragment: lane(l16,half) holds A[M=l16][K=k+koff], [K=k+koff+1]
        v2f a = *(const v2f*)(aptr + k + koff);
        // B 4x16 fragments: lane needs {W[k+koff][n], W[k+koff+1][n]} = sW2[k/2 + half][n]
        const v2f* wrow = sW2 + ((k >> 1) + half) * F;
        v2f b0 = wrow[l16 +  0];
        v2f b1 = wrow[l16 + 16];
        v2f b2 = wrow[l16 + 32];
        v2f b3 = wrow[l16 + 48];
        acc0 = __builtin_amdgcn_wmma_f32_16x16x4_f32(false, a, false, b0, (short)0, acc0, false, false);
        acc1 = __builtin_amdgcn_wmma_f32_16x16x4_f32(false, a, false, b1, (short)0, acc1, false, false);
        acc2 = __builtin_amdgcn_wmma_f32_16x16x4_f32(false, a, false, b2, (short)0, acc2, false, false);
        acc3 = __builtin_amdgcn_wmma_f32_16x16x4_f32(false, a, false, b3, (short)0, acc3, false, false);
    }

    // bias for the 4 column tiles owned by this lane
    float bv0 = sB[l16 +  0], bv1 = sB[l16 + 16], bv2 = sB[l16 + 32], bv3 = sB[l16 + 48];

    // C/D layout: VGPR v of lane -> row (rowBase + v + half*8), col = tile*16 + l16
#pragma unroll
    for (int v = 0; v < 8; ++v) {
        int r = rowBase + v + half * 8;
        if (r < N) {
            float* orow = out + (size_t)r * F;
            orow[l16 +  0] = fmaxf(acc0[v] + bv0, 0.0f);
            orow[l16 + 16] = fmaxf(acc1[v] + bv1, 0.0f);
            orow[l16 + 32] = fmaxf(acc2[v] + bv2, 0.0f);
            orow[l16 + 48] = fmaxf(acc3[v] + bv3, 0.0f);
        }
    }
}

extern "C" void kernel_launch(void* const* d_in, const int* in_sizes, int n_in,
                              void* d_out, int out_size, void* d_ws, size_t ws_size,
                              hipStream_t stream) {
    const float* feat   = (const float*)d_in[0];
    const float* weight = (const float*)d_in[1];
    const float* bias   = (const float*)d_in[2];
    const int*   src    = (const int*)d_in[3];
    const int*   dst    = (const int*)d_in[4];
    float*       out    = (float*)d_out;

    const int N = in_sizes[0] / F;
    const int E = in_sizes[3];

    // workspace layout: agg[N*64] | deg[N] | dinv[N]  (~26.4 MB)
    float* agg  = (float*)d_ws;
    float* deg  = agg + (size_t)N * F;
    float* dinv = deg + N;

    gcn_zero_deg<<<(N + 255) / 256, 256, 0, stream>>>(deg, N);
    gcn_count_deg<<<(E + 255) / 256, 256, 0, stream>>>(dst, deg, E);
    gcn_selfloop<<<(N + 7) / 8, 256, 0, stream>>>(feat, deg, dinv, agg, N);
    gcn_scatter<<<(E + 7) / 8, 256, 0, stream>>>(feat, src, dst, dinv, agg, E);

    int rowTiles = (N + 15) / 16;
    gcn_gemm_relu<<<(rowTiles + 7) / 8, 256, 0, stream>>>(agg, weight, bias, out, N);
}